// GAT_558345748542
// MI455X (gfx1250) — compile-verified
//
#include <hip/hip_runtime.h>
#include <math.h>

typedef __attribute__((ext_vector_type(2))) float v2f;
typedef __attribute__((ext_vector_type(8))) float v8f;
typedef __attribute__((ext_vector_type(4))) unsigned int u32x4;
typedef __attribute__((ext_vector_type(4))) int i32x4;
typedef __attribute__((ext_vector_type(8))) int i32x8;

#define NNODES 50000
#define NEDGES 800000
#define NETOT  (NEDGES + NNODES)
#define FIN    128
#define HIDC   32
#define HEADS  8
#define NCLS   10
#define SLOPE  0.2f

// ---------------------------------------------------------------- utilities

__global__ void fill_kernel(float* __restrict__ p, float v, long long n) {
  long long i = (long long)blockIdx.x * blockDim.x + threadIdx.x;
  if (i < n) p[i] = v;
}

__device__ __forceinline__ void edge_pair(const int* __restrict__ ei,
                                          long long e, int& src, int& dst) {
  if (e < NEDGES) { src = ei[e]; dst = ei[NEDGES + e]; }
  else            { src = dst = (int)(e - NEDGES); }   // self loop
}

__device__ __forceinline__ void atomic_max_f32(float* addr, float val) {
  // sign-split trick: monotone bit patterns, no CAS loop
  if (val >= 0.f) atomicMax((int*)addr, __float_as_int(val));
  else            atomicMin((unsigned int*)addr, (unsigned int)__float_as_int(val));
}

// -------------------------------------------------- TDM: 16 x K f32 tile -> LDS
// Builds a Tensor DMA Descriptor (D#) per CDNA5 ISA §8 and issues
// tensor_load_to_lds. Self-contained tensor == the tile (OOB-safe).
// Called unconditionally: if this toolchain lacks the builtin / wants a
// different arity, the compile error is the probe signal.
__device__ __forceinline__ void tdm_load_tile_16xK(const float* gsrc,
                                                   unsigned lds_off, int K) {
  unsigned long long ga = (unsigned long long)(uintptr_t)gsrc;
  u32x4 g0;
  g0.x = 0x1u;                                            // count=1, load, no gather
  g0.y = lds_off;                                         // lds_addr [63:32]
  g0.z = (unsigned)(ga & 0xffffffffu);                    // global_addr lo
  g0.w = (unsigned)((ga >> 32) & 0x01ffffffu) | (2u << 30); // addr hi + type=2
  i32x8 g1;
  unsigned K16 = (unsigned)K & 0xffffu;
  g1[0] = (int)(2u << 16);                                // data_size = 4 bytes
  g1[1] = (int)(K16 << 16);                               // tensor_dim0[15:0]
  g1[2] = (int)(((unsigned)K >> 16) | (16u << 16));       // dim0 hi | tensor_dim1=16
  g1[3] = (int)(K16 << 16);                               // dim1 hi=0 | tile_dim0=K
  g1[4] = (int)16;                                        // tile_dim1=16, tile_dim2=0
  g1[5] = (int)K;                                         // tensor_dim0_stride lo
  g1[6] = 0;                                              // stride hi | dim1_stride lo
  g1[7] = 0;
  i32x4 z4 = {0, 0, 0, 0};
#if defined(__clang_major__) && __clang_major__ >= 23
  i32x8 z8 = {0, 0, 0, 0, 0, 0, 0, 0};
  __builtin_amdgcn_tensor_load_to_lds(g0, g1, z4, z4, z8, 0);
#else
  __builtin_amdgcn_tensor_load_to_lds(g0, g1, z4, z4, 0);
#endif
  __builtin_amdgcn_s_wait_tensorcnt(0);
}

// ------------------------------------------------------------- WMMA GEMM f32
// C[M x ncols] = A[M x K] * B[K x ncols], all row-major.
// M % 16 == 0, K % 4 == 0, K <= 256. Block = 256 threads (8 waves).
// Block covers 16 rows x 128 cols; each wave owns one 16x16 tile and loops K
// in steps of 4 with v_wmma_f32_16x16x4_f32. A tile staged in LDS by the TDM.
__global__ void wmma_gemm_f32(const float* __restrict__ A,
                              const float* __restrict__ B,
                              float* __restrict__ C,
                              int K, int ncols) {
  __shared__ float As[16 * 256];               // supports K up to 256 (16 KB)
  const int row0 = blockIdx.x * 16;
  const int wave = threadIdx.x >> 5;
  const int lane = threadIdx.x & 31;
  const int col0 = blockIdx.y * 128 + wave * 16;

  // wave 0 DMAs the 16 x K A-tile into LDS via the Tensor Data Mover
  if (wave == 0) {
    tdm_load_tile_16xK(A + (size_t)row0 * K,
                       (unsigned)(uintptr_t)(void*)As, K);
  }
  __syncthreads();

  if (col0 < ncols) {
    const int m = lane & 15;        // A row within tile
    const int g = lane >> 4;        // lane group: K pair {0,1} vs {2,3}
    const int n = col0 + (lane & 15);
    v8f c = {};
    for (int k0 = 0; k0 < K; k0 += 4) {
      v2f a, b;
      a.x = As[m * K + k0 + 2 * g];
      a.y = As[m * K + k0 + 2 * g + 1];
      b.x = B[(size_t)(k0 + 2 * g)     * ncols + n];
      b.y = B[(size_t)(k0 + 2 * g + 1) * ncols + n];
      c = __builtin_amdgcn_wmma_f32_16x16x4_f32(false, a, false, b,
                                                (short)0, c, false, false);
    }
#pragma unroll
    for (int i = 0; i < 8; ++i)     // D: vgpr i -> row i (g=0) / i+8 (g=1)
      C[(size_t)(row0 + i + 8 * g) * ncols + n] = c[i];
  }
}

// --------------------------------------------------- per-node attention dots
// h laid out [N, H, C]; thread t = n*H + hh reads exactly row t*C.
__global__ void alpha_kernel(const float* __restrict__ h,
                             const float* __restrict__ a_s,
                             const float* __restrict__ a_d,
                             float* __restrict__ out_s,
                             float* __restrict__ out_d,
                             int H, int C) {
  int t = blockIdx.x * blockDim.x + threadIdx.x;
  if (t >= NNODES * H) return;
  int hh = t % H;
  const float* row = h + (size_t)t * C;
  float ss = 0.f, dd = 0.f;
  for (int c = 0; c < C; ++c) {
    float v = row[c];
    ss += v * a_s[hh * C + c];
    dd += v * a_d[hh * C + c];
  }
  out_s[t] = ss;
  out_d[t] = dd;
}

// ------------------------------------------------------- edge pass 1: segmax
// One thread per edge; head loop unrolled at compile time.
template <int H>
__global__ void edge_max_kernel(const int* __restrict__ ei,
                                const float* __restrict__ as_,
                                const float* __restrict__ ad_,
                                float* __restrict__ mbuf) {
  long long e = (long long)blockIdx.x * blockDim.x + threadIdx.x;
  if (e >= NETOT) return;
  int src, dst; edge_pair(ei, e, src, dst);
#pragma unroll
  for (int hh = 0; hh < H; ++hh) {
    float v = as_[(size_t)src * H + hh] + ad_[(size_t)dst * H + hh];
    v = v > 0.f ? v : SLOPE * v;                     // leaky_relu
    atomic_max_f32(&mbuf[(size_t)dst * H + hh], v);
  }
}

// --------------------------------------------- edge pass 2: exp + segment sum
template <int H>
__global__ void edge_expsum_kernel(const int* __restrict__ ei,
                                   const float* __restrict__ as_,
                                   const float* __restrict__ ad_,
                                   const float* __restrict__ mbuf,
                                   float* __restrict__ sbuf,
                                   float* __restrict__ exbuf) {
  long long e = (long long)blockIdx.x * blockDim.x + threadIdx.x;
  if (e >= NETOT) return;
  int src, dst; edge_pair(ei, e, src, dst);
#pragma unroll
  for (int hh = 0; hh < H; ++hh) {
    float v = as_[(size_t)src * H + hh] + ad_[(size_t)dst * H + hh];
    v = v > 0.f ? v : SLOPE * v;
    float ex = __expf(v - mbuf[(size_t)dst * H + hh]);
    exbuf[e * H + hh] = ex;
    atomicAdd(&sbuf[(size_t)dst * H + hh], ex);
  }
}

// --------------------------------- edge pass 3: weighted gather / scatter-add
// One wave32 per edge; lane covers channels lane, lane+32, ...
template <int H, int C>
__global__ void edge_agg_kernel(const int* __restrict__ ei,
                                const float* __restrict__ hsrc,
                                const float* __restrict__ exbuf,
                                const float* __restrict__ sbuf,
                                float* __restrict__ outp) {
  const int wave = threadIdx.x >> 5;
  const int lane = threadIdx.x & 31;
  long long e = (long long)blockIdx.x * 8 + wave;
  if (e >= NETOT) return;
  int src, dst; edge_pair(ei, e, src, dst);
  constexpr int Ctot = H * C;
#pragma unroll
  for (int j = 0; j < Ctot / 32; ++j) {
    int ch = lane + 32 * j;
    int hh = ch / C;
    float w = exbuf[e * H + hh] / (sbuf[(size_t)dst * H + hh] + 1e-16f);
    atomicAdd(&outp[(size_t)dst * Ctot + ch],
              hsrc[(size_t)src * Ctot + ch] * w);
  }
}

// --------------------------------------------------------- bias + ELU (inpl)
__global__ void bias_elu_kernel(float* __restrict__ buf,
                                const float* __restrict__ bias, int cmask) {
  long long i = (long long)blockIdx.x * blockDim.x + threadIdx.x;
  if (i >= (long long)NNODES * (cmask + 1)) return;
  float v = buf[i] + bias[(int)i & cmask];
  buf[i] = v > 0.f ? v : (__expf(v) - 1.f);
}

// ------------------------------------------------------------- final linear
__global__ void linear_kernel(const float* __restrict__ h,
                              const float* __restrict__ Wl,
                              const float* __restrict__ bl,
                              float* __restrict__ outp) {
  int t = blockIdx.x * blockDim.x + threadIdx.x;
  if (t >= NNODES * NCLS) return;
  int n = t / NCLS, cc = t % NCLS;
  float acc = bl[cc];
#pragma unroll
  for (int k = 0; k < HIDC; ++k)
    acc += h[(size_t)n * HIDC + k] * Wl[k * NCLS + cc];
  outp[t] = acc;
}

// ---------------------------------------------------------------- dispatcher

static inline int cdiv_ll(long long a, long long b) { return (int)((a + b - 1) / b); }

extern "C" void kernel_launch(void* const* d_in, const int* in_sizes, int n_in,
                              void* d_out, int out_size, void* d_ws, size_t ws_size,
                              hipStream_t stream) {
  const float* x   = (const float*)d_in[0];
  const int*   ei  = (const int*)  d_in[1];
  const float* W1  = (const float*)d_in[2];
  const float* as1 = (const float*)d_in[3];
  const float* ad1 = (const float*)d_in[4];
  const float* b1  = (const float*)d_in[5];
  const float* W2  = (const float*)d_in[6];
  const float* as2 = (const float*)d_in[7];
  const float* ad2 = (const float*)d_in[8];
  const float* b2  = (const float*)d_in[9];
  const float* Wl  = (const float*)d_in[10];
  const float* bl  = (const float*)d_in[11];
  float* out = (float*)d_out;

  // workspace carve-out
  float* p = (float*)d_ws;
  float* h1   = p; p += (size_t)NNODES * 256;   // x @ W1            [N,8,32]
  float* agg1 = p; p += (size_t)NNODES * 256;   // layer-1 output    [N,256]
  float* aS1  = p; p += (size_t)NNODES * HEADS;
  float* aD1  = p; p += (size_t)NNODES * HEADS;
  float* m1   = p; p += (size_t)NNODES * HEADS;
  float* s1   = p; p += (size_t)NNODES * HEADS;
  float* ex1  = p; p += (size_t)NETOT  * HEADS;
  float* h2   = p; p += (size_t)NNODES * HIDC;  // agg1 @ W2         [N,32]
  float* agg2 = p; p += (size_t)NNODES * HIDC;  // layer-2 output    [N,32]
  float* aS2  = p; p += (size_t)NNODES;
  float* aD2  = p; p += (size_t)NNODES;
  float* m2   = p; p += (size_t)NNODES;
  float* s2   = p; p += (size_t)NNODES;
  float* ex2  = p; p += (size_t)NETOT;

  const int TB = 256;
  const float NEG_INF = -INFINITY;

  // ---- init accumulators (ws is poisoned by harness)
  fill_kernel<<<cdiv_ll((long long)NNODES * HEADS, TB), TB, 0, stream>>>(m1, NEG_INF, (long long)NNODES * HEADS);
  fill_kernel<<<cdiv_ll((long long)NNODES * HEADS, TB), TB, 0, stream>>>(s1, 0.f,     (long long)NNODES * HEADS);
  fill_kernel<<<cdiv_ll((long long)NNODES * 256,   TB), TB, 0, stream>>>(agg1, 0.f,   (long long)NNODES * 256);
  fill_kernel<<<cdiv_ll((long long)NNODES,         TB), TB, 0, stream>>>(m2, NEG_INF, (long long)NNODES);
  fill_kernel<<<cdiv_ll((long long)NNODES,         TB), TB, 0, stream>>>(s2, 0.f,     (long long)NNODES);
  fill_kernel<<<cdiv_ll((long long)NNODES * HIDC,  TB), TB, 0, stream>>>(agg2, 0.f,   (long long)NNODES * HIDC);

  // ---- layer 1: h1 = x @ W1  (K=128, 256 cols)
  {
    dim3 grid(NNODES / 16, 256 / 128);
    wmma_gemm_f32<<<grid, TB, 0, stream>>>(x, W1, h1, FIN, HEADS * HIDC);
  }
  alpha_kernel<<<cdiv_ll((long long)NNODES * HEADS, TB), TB, 0, stream>>>(h1, as1, ad1, aS1, aD1, HEADS, HIDC);
  edge_max_kernel<HEADS>   <<<cdiv_ll(NETOT, TB), TB, 0, stream>>>(ei, aS1, aD1, m1);
  edge_expsum_kernel<HEADS><<<cdiv_ll(NETOT, TB), TB, 0, stream>>>(ei, aS1, aD1, m1, s1, ex1);
  edge_agg_kernel<HEADS, HIDC><<<cdiv_ll(NETOT, 8), TB, 0, stream>>>(ei, h1, ex1, s1, agg1);
  bias_elu_kernel<<<cdiv_ll((long long)NNODES * 256, TB), TB, 0, stream>>>(agg1, b1, HEADS * HIDC - 1);

  // ---- layer 2: h2 = agg1 @ W2  (K=256, 32 cols, single head)
  {
    dim3 grid(NNODES / 16, 1);
    wmma_gemm_f32<<<grid, TB, 0, stream>>>(agg1, W2, h2, HEADS * HIDC, HIDC);
  }
  alpha_kernel<<<cdiv_ll(NNODES, TB), TB, 0, stream>>>(h2, as2, ad2, aS2, aD2, 1, HIDC);
  edge_max_kernel<1>   <<<cdiv_ll(NETOT, TB), TB, 0, stream>>>(ei, aS2, aD2, m2);
  edge_expsum_kernel<1><<<cdiv_ll(NETOT, TB), TB, 0, stream>>>(ei, aS2, aD2, m2, s2, ex2);
  edge_agg_kernel<1, HIDC><<<cdiv_ll(NETOT, 8), TB, 0, stream>>>(ei, h2, ex2, s2, agg2);
  bias_elu_kernel<<<cdiv_ll((long long)NNODES * HIDC, TB), TB, 0, stream>>>(agg2, b2, HIDC - 1);

  // ---- classifier
  linear_kernel<<<cdiv_ll((long long)NNODES * NCLS, TB), TB, 0, stream>>>(agg2, Wl, bl, out);

  (void)in_sizes; (void)n_in; (void)out_size; (void)ws_size;
}